// DeMFInet_72292889527094
// MI455X (gfx1250) — compile-verified
//
#include <hip/hip_runtime.h>
#include <hip/hip_bf16.h>

// Problem constants (from setup_inputs): N=8, C=2, H=544, W=960, fp32 throughout.
#define NB  8
#define HH  544
#define WW  960
#define HW  (HH * WW)        // 522240
#define NHW (NB * HW)        // 4177920  (one [N,H,W] plane)

static_assert(NHW % 256 == 0, "splat/combine grids must divide evenly");
static_assert((6 * NHW) % (4 * 256) == 0, "zero grid must divide evenly");

// Workspace layout (floats):
//   f01   : 2*NHW  (warped flow_01 accumulator, [N,2,H,W])
//   f10   : 2*NHW  (warped flow_10 accumulator, [N,2,H,W])
//   norm0 : NHW    (weight accumulator for flow_01; channel-independent)
//   norm1 : NHW    (weight accumulator for flow_10)
// total = 6*NHW floats = ~100.3 MB -> kept resident in MI455X 192 MB L2;
// all one-shot streams (flow reads, output writes) are issued non-temporal.

// ---------------------------------------------------------------------------
// Forward-splat both flow fields. One thread per (n,h,w). Exact grid, no tail.
// ---------------------------------------------------------------------------
__global__ __launch_bounds__(256) void cfr_splat(const float* __restrict__ flow01,
                                                 const float* __restrict__ flow10,
                                                 const float* __restrict__ tval,
                                                 float* __restrict__ f01,
                                                 float* __restrict__ f10,
                                                 float* __restrict__ norm0,
                                                 float* __restrict__ norm1) {
    const int p   = blockIdx.x * 256 + threadIdx.x;     // p < NHW by construction
    const int n   = p / HW;
    const int rem = p - n * HW;
    const int h   = rem / WW;
    const int w   = rem - h * WW;

    const float t = tval[n];

    const size_t cbase = (size_t)n * 2 * HW + rem;  // channel-0 element of this pixel

#pragma unroll
    for (int k = 0; k < 2; ++k) {
        const float* __restrict__ fl = (k == 0) ? flow01 : flow10;
        float*       __restrict__ fd = (k == 0) ? f01    : f10;
        float*       __restrict__ nd = (k == 0) ? norm0  : norm1;
        const float  sc              = (k == 0) ? t      : (1.f - t);

        // img values (also the flow before scaling): ch0 -> y/col shift, ch1 -> x/row shift
        // One-shot streaming reads: non-temporal so they don't evict the accumulators.
        const float v0 = __builtin_nontemporal_load(fl + cbase);       // channel 0
        const float v1 = __builtin_nontemporal_load(fl + cbase + HW);  // channel 1

        const float y = sc * v0;           // column shift
        const float x = sc * v1;           // row shift

        const float x1 = floorf(x);
        const float y1 = floorf(y);
        const float fx = x - x1;           // in [0,1)
        const float fy = y - y1;

        // exp(-(a^2 + b^2)) = exp(-a^2) * exp(-b^2): 4 v_exp_f32, reused across corners
        const float ex1 = __expf(-fx * fx);
        const float ex2 = __expf(-(fx - 1.f) * (fx - 1.f));
        const float ey1 = __expf(-fy * fy);
        const float ey2 = __expf(-(fy - 1.f) * (fy - 1.f));

        const int ix1 = h + (int)x1;       // row target (corner 1)
        const int iy1 = w + (int)y1;       // col target (corner 1)

        float* __restrict__ dch0 = fd + (size_t)n * 2 * HW;  // channel-0 plane
        float* __restrict__ dnrm = nd + (size_t)n * HW;      // norm plane

        const int base = ix1 * WW + iy1;   // corner (0,0) offset; others are +1, +WW, +WW+1

#pragma unroll
        for (int cx = 0; cx < 2; ++cx) {
#pragma unroll
            for (int cy = 0; cy < 2; ++cy) {
                const int ix = ix1 + cx;
                const int iy = iy1 + cy;
                if (ix >= 0 && ix < HH && iy >= 0 && iy < WW) {
                    const float wg  = (cx ? ex2 : ex1) * (cy ? ey2 : ey1);
                    const int   off = base + cx * WW + cy;
                    // no-return float atomics -> global_atomic_add_f32 (STOREcnt path)
                    atomicAdd(dch0 + off,      v0 * wg);
                    atomicAdd(dch0 + HW + off, v1 * wg);
                    atomicAdd(dnrm + off,      wg);
                }
            }
        }
    }
}

// ---------------------------------------------------------------------------
// Zero the 100 MB accumulator scratch (poisoned by harness; re-zero each call).
// ---------------------------------------------------------------------------
__global__ __launch_bounds__(256) void cfr_zero_ws(float4* __restrict__ ws) {
    const int i = blockIdx.x * 256 + threadIdx.x;       // exact grid
    ws[i] = make_float4(0.f, 0.f, 0.f, 0.f);
}

// ---------------------------------------------------------------------------
// Normalize + blend into flow_t0 / flow_t1. One thread per (n,h,w), both chans.
// ---------------------------------------------------------------------------
__global__ __launch_bounds__(256) void cfr_combine(const float* __restrict__ f01,
                                                   const float* __restrict__ f10,
                                                   const float* __restrict__ norm0,
                                                   const float* __restrict__ norm1,
                                                   const float* __restrict__ tval,
                                                   float* __restrict__ out0,
                                                   float* __restrict__ out1) {
    const int p   = blockIdx.x * 256 + threadIdx.x;     // exact grid
    const int n   = p / HW;
    const int rem = p - n * HW;

    const float t = tval[n];
    const float u = 1.f - t;

    // Last use of the norm planes: non-temporal read.
    const float nrm = u * __builtin_nontemporal_load(norm0 + p)
                    + t * __builtin_nontemporal_load(norm1 + p);
    // mask math: mask ? v/norm : v   ==  v * (norm>0 ? 1/norm : 1)
    const float inv = (nrm > 0.f) ? (1.f / nrm) : 1.f;

    const float c00 = -u * t;   // flow_t0: f01 coeff
    const float c01 = t * t;    // flow_t0: f10 coeff
    const float c10 = u * u;    // flow_t1: f01 coeff
    const float c11 = -t * u;   // flow_t1: f10 coeff

    const size_t b = (size_t)n * 2 * HW + rem;
#pragma unroll
    for (int c = 0; c < 2; ++c) {
        const size_t idx = b + (size_t)c * HW;
        const float a0 = __builtin_nontemporal_load(f01 + idx);  // last use
        const float a1 = __builtin_nontemporal_load(f10 + idx);  // last use
        // Write-once outputs: non-temporal stores, keep L2 for the accumulators.
        __builtin_nontemporal_store((c00 * a0 + c01 * a1) * inv, out0 + idx);
        __builtin_nontemporal_store((c10 * a0 + c11 * a1) * inv, out1 + idx);
    }
}

extern "C" void kernel_launch(void* const* d_in, const int* in_sizes, int n_in,
                              void* d_out, int out_size, void* d_ws, size_t ws_size,
                              hipStream_t stream) {
    (void)in_sizes; (void)n_in; (void)out_size; (void)ws_size;

    const float* flow01 = (const float*)d_in[0];   // [8,2,544,960] f32
    const float* flow10 = (const float*)d_in[1];   // [8,2,544,960] f32
    const float* tval   = (const float*)d_in[2];   // [8,1,1,1]     f32

    float* ws    = (float*)d_ws;
    float* f01   = ws;                        // 2*NHW
    float* f10   = ws + 2 * (size_t)NHW;      // 2*NHW
    float* norm0 = ws + 4 * (size_t)NHW;      // NHW
    float* norm1 = ws + 5 * (size_t)NHW;      // NHW

    float* out0 = (float*)d_out;                    // flow_t0 [8,2,544,960]
    float* out1 = (float*)d_out + 2 * (size_t)NHW;  // flow_t1 [8,2,544,960]

    const int threads = 256;

    // 1) zero the accumulators (exact grid: 6*NHW/4 float4 stores)
    cfr_zero_ws<<<(6 * NHW) / (4 * threads), threads, 0, stream>>>((float4*)ws);

    // 2) forward splat (float atomics into L2-resident accumulators)
    cfr_splat<<<NHW / threads, threads, 0, stream>>>(
        flow01, flow10, tval, f01, f10, norm0, norm1);

    // 3) normalize + blend into the two outputs (NT streams out)
    cfr_combine<<<NHW / threads, threads, 0, stream>>>(
        f01, f10, norm0, norm1, tval, out0, out1);
}